// GraphNN_68281390072484
// MI455X (gfx1250) — compile-verified
//
#include <hip/hip_runtime.h>

typedef float v2f __attribute__((ext_vector_type(2)));
typedef float v8f __attribute__((ext_vector_type(8)));

// ---------------------------------------------------------------------------
// C[M x N] = A[M x K] @ B[K x N], all row-major f32. M tiles of 16, N%32==0,
// K%4==0, K<=128. One block = 8 waves sharing one 32-column B strip staged in
// LDS; each wave computes a 16x32 strip (two 16x16 WMMA tiles, shared A frag).
// Uses V_WMMA_F32_16X16X4_F32 (full f32 precision on the matrix pipe).
//
// LDS layout: B[k][c] -> bs[(k>>1)*96 + c*2 + (k&1)]
//   * (k,k+1) adjacent -> B fragment is one ds_load_b64 per lane
//   * 96-float pitch (384B = 32-bank shift) -> the kh=0 and kh=2 half-waves
//     hit disjoint bank halves: conflict-free b64 reads
// ---------------------------------------------------------------------------
__global__ __launch_bounds__(256) void gcn_gemm_wmma_f32(
    const float* __restrict__ A, const float* __restrict__ B,
    float* __restrict__ C, int M, int N, int K)
{
  __shared__ float bs[64 * 96];                   // 24 KB (K<=128, 32 cols)

  int npairs = N >> 5;                            // 32-col strips
  int mg = blockIdx.x / npairs;                   // group of 8 M-tiles
  int np = blockIdx.x % npairs;
  int col0 = np << 5;

  // cooperative load of the K x 32 B strip into LDS
  for (int i = threadIdx.x; i < K * 32; i += 256) {
    int k = i >> 5;
    int c = i & 31;
    bs[(k >> 1) * 96 + c * 2 + (k & 1)] = B[(long long)k * N + col0 + c];
  }
  __syncthreads();

  int wv   = threadIdx.x >> 5;
  int lane = threadIdx.x & 31;
  int mt   = mg * 8 + wv;
  int mtiles = M >> 4;
  if (mt < mtiles) {                              // wave-uniform: EXEC all-1s inside
    int m  = (mt << 4) + (lane & 15);
    int kh = (lane >> 4) << 1;                    // 0 or 2
    const float* arow = A + (long long)m * K + kh;
    const float* bsl0 = bs + (lane & 15) * 2;
    const float* bsl1 = bs + ((lane & 15) + 16) * 2;

    v8f acc0 = {};
    v8f acc1 = {};
    for (int k = 0; k < K; k += 4) {
      v2f a  = *(const v2f*)(arow + k);           // contiguous pair, 8B aligned
      int p  = (k + kh) >> 1;
      v2f b0 = *(const v2f*)(bsl0 + p * 96);      // ds_load_b64
      v2f b1 = *(const v2f*)(bsl1 + p * 96);      // ds_load_b64
      acc0 = __builtin_amdgcn_wmma_f32_16x16x4_f32(false, a, false, b0, (short)0, acc0, false, false);
      acc1 = __builtin_amdgcn_wmma_f32_16x16x4_f32(false, a, false, b1, (short)0, acc1, false, false);
    }

    int crow = (mt << 4) + ((lane >> 4) << 3);
    float* cp = C + (long long)crow * N + col0 + (lane & 15);
#pragma unroll
    for (int v = 0; v < 8; ++v) {
      cp[(long long)v * N]      = acc0[v];
      cp[(long long)v * N + 16] = acc1[v];
    }
  }
}

// ---------------------------------------------------------------------------
// Degree / normalization
// ---------------------------------------------------------------------------
__global__ void gcn_init_deg(float* __restrict__ deg, int n) {
  int i = blockIdx.x * blockDim.x + threadIdx.x;
  if (i < n) deg[i] = 1.0f;                       // self-loop contributes 1
}

__global__ void gcn_accum_deg(const long long* __restrict__ dst,
                              float* __restrict__ deg, long long E) {
  long long e = (long long)blockIdx.x * blockDim.x + threadIdx.x;
  if (e < E) atomicAdd(&deg[(int)dst[e]], 1.0f);
}

__global__ void gcn_deg_to_dinv(float* __restrict__ deg, int n) {
  int i = blockIdx.x * blockDim.x + threadIdx.x;
  if (i < n) deg[i] = rsqrtf(deg[i]);             // deg >= 1 always (self-loops)
}

// ---------------------------------------------------------------------------
// Edge scatter: agg[dst] += h[src] * (dinv[src]*dinv[dst])
// thread = (edge, group of 4 features); h row read is one global_load_b128.
// At D=128 one wave covers one edge -> fully coalesced 512B gather.
// ---------------------------------------------------------------------------
__global__ void gcn_scatter_edges(const long long* __restrict__ src,
                                  const long long* __restrict__ dst,
                                  const float* __restrict__ h,
                                  const float* __restrict__ dinv,
                                  float* __restrict__ agg,
                                  long long E, int D) {
  const int FPT = 4;
  long long tid = (long long)blockIdx.x * blockDim.x + threadIdx.x;
  int groups = D / FPT;
  long long e = tid / groups;
  int g = (int)(tid % groups);
  if (e >= E) return;
  int s = (int)src[e];
  int d = (int)dst[e];
  float norm = dinv[s] * dinv[d];
  const float4 hv = *(const float4*)(h + (long long)s * D + g * FPT);
  float* ap = agg + (long long)d * D + g * FPT;
  atomicAdd(ap + 0, hv.x * norm);
  atomicAdd(ap + 1, hv.y * norm);
  atomicAdd(ap + 2, hv.z * norm);
  atomicAdd(ap + 3, hv.w * norm);
}

// ---------------------------------------------------------------------------
// Self-loop contribution + bias (+ optional ReLU):
// out[n,f] = act(agg[n,f] + h[n,f]*dinv[n]^2 + bias[f])
// No atomics needed: stream ordering guarantees the scatter has completed.
// ---------------------------------------------------------------------------
__global__ void gcn_finish_layer(const float* __restrict__ h,
                                 const float* __restrict__ agg,
                                 const float* __restrict__ dinv,
                                 const float* __restrict__ bias,
                                 float* __restrict__ out,
                                 int n, int D, int do_relu) {
  long long idx = (long long)blockIdx.x * blockDim.x + threadIdx.x;
  if (idx >= (long long)n * D) return;
  int node = (int)(idx / D);
  int f    = (int)(idx % D);
  float w = dinv[node];
  float v = agg[idx] + h[idx] * (w * w) + bias[f];
  out[idx] = do_relu ? fmaxf(v, 0.0f) : v;
}

// ---------------------------------------------------------------------------
// In-place log_softmax over 64 columns: one wave per node, 2 elems per lane,
// wave32 shuffle reductions.
// ---------------------------------------------------------------------------
__global__ __launch_bounds__(256) void gcn_log_softmax64(float* __restrict__ out, int n) {
  int wave = (int)((blockIdx.x * (long long)blockDim.x + threadIdx.x) >> 5);
  int lane = threadIdx.x & 31;
  if (wave >= n) return;
  float* row = out + (long long)wave * 64;
  float a = row[lane];
  float b = row[lane + 32];
  float m = fmaxf(a, b);
#pragma unroll
  for (int off = 16; off > 0; off >>= 1) m = fmaxf(m, __shfl_xor(m, off, 32));
  float se = expf(a - m) + expf(b - m);
#pragma unroll
  for (int off = 16; off > 0; off >>= 1) se += __shfl_xor(se, off, 32);
  float lse = m + logf(se);
  row[lane]      = a - lse;
  row[lane + 32] = b - lse;
}

// ---------------------------------------------------------------------------
extern "C" void kernel_launch(void* const* d_in, const int* in_sizes, int n_in,
                              void* d_out, int out_size, void* d_ws, size_t ws_size,
                              hipStream_t stream) {
  const float*     x   = (const float*)d_in[0];
  const long long* ei  = (const long long*)d_in[1];   // int64 [2, E]
  const float*     W1  = (const float*)d_in[2];
  const float*     b1  = (const float*)d_in[3];
  const float*     W2  = (const float*)d_in[4];
  const float*     b2  = (const float*)d_in[5];
  float*           out = (float*)d_out;

  const int IN  = 128, HID = 128, OUT = 64;
  const int       Nn = in_sizes[0] / IN;     // 50000 (== 3125 * 16, tiles exact)
  const long long E  = in_sizes[1] / 2;      // 800000
  const long long* srcI = ei;
  const long long* dstI = ei + E;

  // workspace layout (f32): dinv[Nn] | bufA[Nn*128] | bufB[Nn*128]
  float* dinv  = (float*)d_ws;
  float* bufA  = dinv + Nn;                               // h, later h2raw + agg2
  float* bufB  = bufA + (long long)Nn * HID;              // agg1 -> h1 (in place)
  float* h2raw = bufA;                                    // Nn*64, reuses bufA
  float* agg2  = bufA + (long long)Nn * OUT;              // Nn*64, second half of bufA

  const int T = 256;
  const int mtiles = Nn / 16;
  const int mgroups = (mtiles + 7) / 8;

  // 1) degree -> dinv
  gcn_init_deg<<<(Nn + T - 1) / T, T, 0, stream>>>(dinv, Nn);
  gcn_accum_deg<<<(unsigned)((E + T - 1) / T), T, 0, stream>>>(dstI, dinv, E);
  gcn_deg_to_dinv<<<(Nn + T - 1) / T, T, 0, stream>>>(dinv, Nn);

  // 2) h = x @ W1  (WMMA f32, LDS-staged B strip)
  gcn_gemm_wmma_f32<<<mgroups * (HID / 32), T, 0, stream>>>(x, W1, bufA, Nn, HID, IN);

  // 3) agg1 = 0; scatter edges; add self-loop + bias + ReLU -> h1 (bufB)
  hipMemsetAsync(bufB, 0, (size_t)Nn * HID * sizeof(float), stream);
  {
    long long threads = E * (HID / 4);
    gcn_scatter_edges<<<(unsigned)((threads + T - 1) / T), T, 0, stream>>>(
        srcI, dstI, bufA, dinv, bufB, E, HID);
  }
  {
    long long threads = (long long)Nn * HID;
    gcn_finish_layer<<<(unsigned)((threads + T - 1) / T), T, 0, stream>>>(
        bufA, bufB, dinv, b1, bufB, Nn, HID, 1);
  }

  // 4) h2raw = h1 @ W2  (WMMA f32, LDS-staged B strip)
  gcn_gemm_wmma_f32<<<mgroups * (OUT / 32), T, 0, stream>>>(bufB, W2, h2raw, Nn, OUT, HID);

  // 5) agg2 = 0; scatter edges (D=64); finish into d_out (no relu)
  hipMemsetAsync(agg2, 0, (size_t)Nn * OUT * sizeof(float), stream);
  {
    long long threads = E * (OUT / 4);
    gcn_scatter_edges<<<(unsigned)((threads + T - 1) / T), T, 0, stream>>>(
        srcI, dstI, h2raw, dinv, agg2, E, OUT);
  }
  {
    long long threads = (long long)Nn * OUT;
    gcn_finish_layer<<<(unsigned)((threads + T - 1) / T), T, 0, stream>>>(
        h2raw, agg2, dinv, b2, out, Nn, OUT, 0);
  }

  // 6) in-place log_softmax over 64 cols, one wave per node
  {
    long long threads = (long long)Nn * 32;
    gcn_log_softmax64<<<(unsigned)((threads + T - 1) / T), T, 0, stream>>>(out, Nn);
  }
}